// NoisyTopkRouter_8461085573276
// MI455X (gfx1250) — compile-verified
//
#include <hip/hip_runtime.h>
#include <math.h>

// NoisyTopkRouter for MI455X (gfx1250, wave32).
// Fused GEMM [32768 x 1824] x [1824 x 32] via V_WMMA_F32_16X16X4_F32.
// Per wave: M=32 (two 16-row tiles) so each ds_load_2addr_b64 B-fetch feeds
// 4 WMMAs. Weights pre-swizzled into 228 KB LDS (320 KB/WGP on CDNA5).
// Epilogue (softplus noise, softmax, top-2 masked softmax) in registers via
// 16-lane shuffles.

typedef __attribute__((ext_vector_type(2))) float v2f;
typedef __attribute__((ext_vector_type(8))) float v8f;

#define B_      8
#define T_      4096
#define NE_     1024
#define E_      16
#define LIN_    1824
#define NSTEPS  (LIN_ / 4)          // 456 K-steps of 4
#define TOKENS  (B_ * T_)           // 32768
#define SMEM_BYTES (NSTEPS * 64 * (int)sizeof(v2f))  // [step][mat(2)][lane(32)] float2 = 228 KB

// max+argmax over the 16-lane half-wave (xor masks <= 15 stay in-group)
__device__ __forceinline__ void half16_max_arg(float& v, int& idx) {
#pragma unroll
    for (int m = 8; m >= 1; m >>= 1) {
        float ov = __shfl_xor(v, m, 32);
        int   oi = __shfl_xor(idx, m, 32);
        if (ov > v || (ov == v && oi < idx)) { v = ov; idx = oi; }
    }
}

__device__ __forceinline__ float half16_sum(float v) {
#pragma unroll
    for (int m = 8; m >= 1; m >>= 1) v += __shfl_xor(v, m, 32);
    return v;
}

// Epilogue for one 16-token tile held in (accR, accN), C/D layout:
// VGPR r: lanes 0-15 -> M=r, lanes 16-31 -> M=r+8; N = lane&15.
__device__ __forceinline__ void epilogue_tile(
    const v8f& accR, const v8f& accN, int m0, int n, int hi,
    float brt, float bns,
    const float* __restrict__ noise_eps,
    float* __restrict__ out_router,
    float* __restrict__ out_indices,
    float* __restrict__ out_gate)
{
#pragma unroll
    for (int r = 0; r < 8; ++r) {
        const int   tokR  = m0 + r + hi * 8;
        const float logit = accR[r] + brt;
        const float nz    = accN[r] + bns;
        // numerically stable softplus
        const float sp    = fmaxf(nz, 0.f) + log1pf(expf(-fabsf(nz)));
        const float ev    = noise_eps[(size_t)tokR * E_ + n];
        const float noisy = logit + ev * sp;

        // top-1 over the 16 experts of this row
        float v1 = noisy; int i1 = n;
        half16_max_arg(v1, i1);
        // top-2
        float v2 = (n == i1) ? -INFINITY : noisy; int i2 = n;
        half16_max_arg(v2, i2);

        // dense softmax (gate1)
        const float ex   = expf(noisy - v1);
        const float ssum = half16_sum(ex);
        const float gate = ex / ssum;

        // top-2 masked softmax (router_output)
        const float denom = 1.f + expf(v2 - v1);
        const bool  in2   = (n == i1) || (n == i2);
        const float rv    = in2 ? (ex / denom) : 0.f;

        const size_t o = (size_t)tokR * E_ + n;
        out_router[o] = rv;
        out_gate[o]   = gate;
        if (n == 0) {
            out_indices[(size_t)tokR * 2 + 0] = (float)i1;
            out_indices[(size_t)tokR * 2 + 1] = (float)i2;
        }
    }
}

__global__ void noisy_topk_router_kernel(
    const float* __restrict__ mh,        // [TOKENS, 1024]
    const float* __restrict__ dt,        // [TOKENS, 256]
    const float* __restrict__ dd,        // [TOKENS, 256]
    const float* __restrict__ dr,        // [TOKENS, 128]
    const float* __restrict__ de,        // [TOKENS, 128]
    const float* __restrict__ city,      // [4, 32]
    const float* __restrict__ w_route,   // [1824, 16] row-major
    const float* __restrict__ b_route,   // [16]
    const float* __restrict__ w_noise,   // [1824, 16]
    const float* __restrict__ b_noise,   // [16]
    const float* __restrict__ noise_eps, // [TOKENS, 16]
    const int*  __restrict__ city_index, // scalar
    float* __restrict__ out_router,      // [TOKENS, 16]
    float* __restrict__ out_indices,     // [TOKENS, 2] (as float)
    float* __restrict__ out_gate)        // [TOKENS, 16]
{
    extern __shared__ v2f smem[];        // [NSTEPS][2][32] in B-matrix lane layout

    const int tid  = threadIdx.x;
    const int lane = tid & 31;
    const int wave = tid >> 5;

    // ---- Stage both weight matrices into LDS, pre-swizzled to the 4x16 f32
    // B-matrix register layout: VGPR0 = rows K+0 (lanes 0-15) / K+2 (lanes 16-31),
    // VGPR1 = rows K+1 / K+3; N = lane&15.
    for (int idx = tid; idx < NSTEPS * 64; idx += blockDim.x) {
        const int s   = idx >> 6;
        const int rem = idx & 63;
        const int mat = rem >> 5;
        const int l   = rem & 31;
        const int nn  = l & 15;
        const int kb  = s * 4 + ((l >> 4) << 1);
        const float* w = mat ? w_noise : w_route;
        v2f v;
        v.x = w[kb * E_ + nn];
        v.y = w[(kb + 1) * E_ + nn];
        smem[idx] = v;
    }
    __syncthreads();

    // ---- Two 16-token tiles (M=32) per wave ----
    const int tile = blockIdx.x * (blockDim.x >> 5) + wave;
    const int m0   = tile * 32;
    if (m0 >= TOKENS) return;

    const int n      = lane & 15;   // A-row token / output expert column
    const int hi     = lane >> 4;   // half-wave select
    const int klocal = hi * 2;      // lanes 16-31 hold K+2, K+3
    const int tok0   = m0 + n;      // tile 0 row
    const int tok1   = m0 + 16 + n; // tile 1 row

    const float* cityp = city + (*city_index) * 32;

    const float* base0[6] = {
        mh + (size_t)tok0 * NE_ + klocal,
        cityp + klocal,                      // broadcast segment (stride 0)
        dt + (size_t)tok0 * 256 + klocal,
        dd + (size_t)tok0 * 256 + klocal,
        dr + (size_t)tok0 * 128 + klocal,
        de + (size_t)tok0 * 128 + klocal
    };
    const float* base1[6] = {
        mh + (size_t)tok1 * NE_ + klocal,
        cityp + klocal,
        dt + (size_t)tok1 * 256 + klocal,
        dd + (size_t)tok1 * 256 + klocal,
        dr + (size_t)tok1 * 128 + klocal,
        de + (size_t)tok1 * 128 + klocal
    };
    const int lens[6] = { 1024, 32, 256, 256, 128, 128 };

    v8f acc0R = {0.f, 0.f, 0.f, 0.f, 0.f, 0.f, 0.f, 0.f};
    v8f acc0N = acc0R;
    v8f acc1R = acc0R;
    v8f acc1N = acc0R;

    int s = 0;
    for (int seg = 0; seg < 6; ++seg) {
        const float* a0  = base0[seg];
        const float* a1  = base1[seg];
        const int    len = lens[seg];
        // 32-float chunks: one unconditional prefetch, 8 fully-unrolled K4
        // steps -> 32 WMMAs per iteration with minimal scalar overhead.
        for (int kc = 0; kc < len; kc += 32) {
            __builtin_prefetch(a0 + kc + 256, 0, 3);  // global_prefetch_b8
            __builtin_prefetch(a1 + kc + 256, 0, 3);
            const v2f* bp = smem + s * 64 + lane;
#pragma unroll
            for (int k = 0; k < 32; k += 4) {
                v2f av0 = *(const v2f*)(a0 + kc + k);  // 16x4 f32 A tile, b64 load
                v2f av1 = *(const v2f*)(a1 + kc + k);
                v2f br  = bp[(k >> 2) * 64];           // 4x16 f32 B (route)
                v2f bn  = bp[(k >> 2) * 64 + 32];      // 4x16 f32 B (noise)
                acc0R = __builtin_amdgcn_wmma_f32_16x16x4_f32(
                    false, av0, false, br, (short)0, acc0R, false, false);
                acc0N = __builtin_amdgcn_wmma_f32_16x16x4_f32(
                    false, av0, false, bn, (short)0, acc0N, false, false);
                acc1R = __builtin_amdgcn_wmma_f32_16x16x4_f32(
                    false, av1, false, br, (short)0, acc1R, false, false);
                acc1N = __builtin_amdgcn_wmma_f32_16x16x4_f32(
                    false, av1, false, bn, (short)0, acc1N, false, false);
            }
            s += 8;
        }
    }

    // ---- Epilogue ----
    const float brt = b_route[n];
    const float bns = b_noise[n];

    epilogue_tile(acc0R, acc0N, m0,      n, hi, brt, bns,
                  noise_eps, out_router, out_indices, out_gate);
    epilogue_tile(acc1R, acc1N, m0 + 16, n, hi, brt, bns,
                  noise_eps, out_router, out_indices, out_gate);
}

extern "C" void kernel_launch(void* const* d_in, const int* in_sizes, int n_in,
                              void* d_out, int out_size, void* d_ws, size_t ws_size,
                              hipStream_t stream) {
    const float* mh    = (const float*)d_in[0];
    const float* dt    = (const float*)d_in[1];
    const float* dd    = (const float*)d_in[2];
    const float* dr    = (const float*)d_in[3];
    const float* de    = (const float*)d_in[4];
    const float* city  = (const float*)d_in[5];
    const float* w_r   = (const float*)d_in[6];
    const float* b_r   = (const float*)d_in[7];
    const float* w_n   = (const float*)d_in[8];
    const float* b_n   = (const float*)d_in[9];
    const float* eps   = (const float*)d_in[10];
    const int*   cidx  = (const int*)d_in[11];

    float* out_router  = (float*)d_out;                       // [TOKENS*16]
    float* out_indices = out_router + (size_t)TOKENS * E_;    // [TOKENS*2]
    float* out_gate    = out_indices + (size_t)TOKENS * 2;    // [TOKENS*16]

    (void)hipFuncSetAttribute((const void*)noisy_topk_router_kernel,
                              hipFuncAttributeMaxDynamicSharedMemorySize,
                              SMEM_BYTES);

    const int threads = 256;                      // 8 waves (wave32)
    const int tiles   = TOKENS / 32;              // 1024 (M=32 per wave)
    const int blocks  = tiles / (threads / 32);   // 128

    noisy_topk_router_kernel<<<blocks, threads, SMEM_BYTES, stream>>>(
        mh, dt, dd, dr, de, city, w_r, b_r, w_n, b_n, eps, cidx,
        out_router, out_indices, out_gate);
}